// AV_loss_34969623724104
// MI455X (gfx1250) — compile-verified
//
#include <hip/hip_runtime.h>
#include <math.h>

typedef __attribute__((ext_vector_type(2))) float v2f;
typedef __attribute__((ext_vector_type(8))) float v8f;

#define NB 1000      // main-kernel blocks: NB*NT threads == 256000 quads exactly
#define NT 256       // 8 waves of 32

// ---------------------------------------------------------------------------
// Wave32 sum via V_WMMA_F32_16X16X4_F32.
// A: documented layout (lane m<16: vgpr0=K0, vgpr1=K1; lane m+16: vgpr0=K2,
//    vgpr1=K3). We set vgpr0 = x, vgpr1 = 0  =>  row m of A = [x[m],0,x[m+16],0].
// B: ALL ONES -> layout-independent: D[m][n] = x[m] + x[m+16] for every n.
// D: documented layout (vgpr r, lanes 0-15 -> M=r; lanes 16-31 -> M=r+8).
// Per-lane sum of the 8 D vgprs gives sum(rows 0..7) on lanes 0-15 and
// sum(rows 8..15) on lanes 16-31; shfl_xor(16) completes the wave sum.
// ---------------------------------------------------------------------------
__device__ __forceinline__ float wave_sum32(float x) {
  v2f a; a[0] = x;    a[1] = 0.0f;
  v2f b; b[0] = 1.0f; b[1] = 1.0f;
  v8f c = {};
  v8f d = __builtin_amdgcn_wmma_f32_16x16x4_f32(
      /*neg_a=*/false, a, /*neg_b=*/false, b,
      /*c_mod=*/(short)0, c, /*reuse_a=*/false, /*reuse_b=*/false);
  float s = d[0] + d[1] + d[2] + d[3] + d[4] + d[5] + d[6] + d[7];
  s += __shfl_xor(s, 16, 32);
  return s;   // full wave sum in every lane
}

// ---------------------------------------------------------------------------
// Closed-form symmetric 3x3 eigensolver.
// Entries: [a b c; b d e; c e f].  Outputs ascending eigenvalues w0<=w1<=w2
// and the unit eigenvector of w2 (sign-ambiguous; caller takes |dot|).
// ---------------------------------------------------------------------------
__device__ __forceinline__ void eig3(float a, float b, float c,
                                     float d, float e, float f,
                                     float& w0, float& w1, float& w2,
                                     float& vx, float& vy, float& vz) {
  const float q  = (a + d + f) * (1.0f / 3.0f);
  const float p1 = b * b + c * c + e * e;
  const float aa = a - q, dd = d - q, ff = f - q;
  const float p2 = aa * aa + dd * dd + ff * ff + 2.0f * p1;
  float p = sqrtf(p2 * (1.0f / 6.0f));
  p = fmaxf(p, 1e-30f);
  const float ip = 1.0f / p;
  const float b00 = aa * ip, b01 = b * ip, b02 = c * ip;
  const float b11 = dd * ip, b12 = e * ip, b22 = ff * ip;
  float r = 0.5f * (b00 * (b11 * b22 - b12 * b12)
                  - b01 * (b01 * b22 - b12 * b02)
                  + b02 * (b01 * b12 - b11 * b02));
  r = fminf(fmaxf(r, -1.0f), 1.0f);
  const float phi = acosf(r) * (1.0f / 3.0f);
  const float TWO_PI_3 = 2.09439510239319549f;
  w2 = q + 2.0f * p * cosf(phi);             // largest
  w0 = q + 2.0f * p * cosf(phi + TWO_PI_3);  // smallest
  w1 = 3.0f * q - w0 - w2;                   // middle (trace identity)

  // Principal eigenvector: null space of (A - w2*I) via row cross products.
  const float r0x = a - w2, r0y = b,      r0z = c;
  const float r1x = b,      r1y = d - w2, r1z = e;
  const float r2x = c,      r2y = e,      r2z = f - w2;
  const float c0x = r0y * r1z - r0z * r1y, c0y = r0z * r1x - r0x * r1z, c0z = r0x * r1y - r0y * r1x;
  const float c1x = r0y * r2z - r0z * r2y, c1y = r0z * r2x - r0x * r2z, c1z = r0x * r2y - r0y * r2x;
  const float c2x = r1y * r2z - r1z * r2y, c2y = r1z * r2x - r1x * r2z, c2z = r1x * r2y - r1y * r2x;
  const float n0 = c0x * c0x + c0y * c0y + c0z * c0z;
  const float n1 = c1x * c1x + c1y * c1y + c1z * c1z;
  const float n2 = c2x * c2x + c2y * c2y + c2z * c2z;
  float best = n0, bx = c0x, by = c0y, bz = c0z;
  if (n1 > best) { best = n1; bx = c1x; by = c1y; bz = c1z; }
  if (n2 > best) { best = n2; bx = c2x; by = c2y; bz = c2z; }
  const float inv = rsqrtf(fmaxf(best, 1e-38f));
  vx = bx * inv; vy = by * inv; vz = bz * inv;
}

// ---------------------------------------------------------------------------
// Main kernel: each thread handles 4 consecutive voxels (float4/int4 loads),
// accumulates (sum|dw|, sum|dot|, count), reduces wave->block, writes block
// partials to workspace. Deterministic, no atomics.
// ---------------------------------------------------------------------------
__global__ __launch_bounds__(NT)
void av_loss_main(const float* __restrict__ in, const float* __restrict__ tg,
                  const int* __restrict__ mask, float* __restrict__ part) {
  constexpr int S = 80 * 80 * 80;   // 512000 voxels per batch image
  constexpr int BSTRIDE = 6 * S;    // batch stride in input/target

  const int t  = blockIdx.x * NT + threadIdx.x;  // quad index, [0, 256000)
  const int v0 = t * 4;
  const int b  = v0 / S;            // 512000 % 4 == 0: quads never cross batches
  const int s  = v0 - b * S;

  const float* ipb = in + b * BSTRIDE + s;
  const float* tpb = tg + b * BSTRIDE + s;
  const int4 mk = *(const int4*)(mask + b * S + s);

  float4 ic[6], tc[6];
#pragma unroll
  for (int ch = 0; ch < 6; ++ch) {
    ic[ch] = *(const float4*)(ipb + ch * S);
    tc[ch] = *(const float4*)(tpb + ch * S);
  }
  const int mv[4] = {mk.x, mk.y, mk.z, mk.w};

  float lval = 0.0f, ldot = 0.0f, lcnt = 0.0f;
#pragma unroll
  for (int j = 0; j < 4; ++j) {
    const float mf = (mv[j] == 1) ? 1.0f : 0.0f;  // multiply, not branch: keeps EXEC full
    const float* icf[6] = {(const float*)&ic[0], (const float*)&ic[1], (const float*)&ic[2],
                           (const float*)&ic[3], (const float*)&ic[4], (const float*)&ic[5]};
    const float* tcf[6] = {(const float*)&tc[0], (const float*)&tc[1], (const float*)&tc[2],
                           (const float*)&tc[3], (const float*)&tc[4], (const float*)&tc[5]};
    float wi0, wi1, wi2, vix, viy, viz;
    float wt0, wt1, wt2, vtx, vty, vtz;
    eig3(icf[0][j], icf[1][j], icf[2][j], icf[3][j], icf[4][j], icf[5][j],
         wi0, wi1, wi2, vix, viy, viz);
    eig3(tcf[0][j], tcf[1][j], tcf[2][j], tcf[3][j], tcf[4][j], tcf[5][j],
         wt0, wt1, wt2, vtx, vty, vtz);
    lval += mf * (fabsf(wi0 - wt0) + fabsf(wi1 - wt1) + fabsf(wi2 - wt2));
    ldot += mf * fabsf(vix * vtx + viy * vty + viz * vtz);
    lcnt += mf;
  }

  // Wave reduction via WMMA, then cross-wave via LDS.
  const float wv = wave_sum32(lval);
  const float wd = wave_sum32(ldot);
  const float wc = wave_sum32(lcnt);

  __shared__ float sv[NT / 32][3];
  const int lane = threadIdx.x & 31;
  const int wave = threadIdx.x >> 5;
  if (lane == 0) { sv[wave][0] = wv; sv[wave][1] = wd; sv[wave][2] = wc; }
  __syncthreads();
  if (threadIdx.x == 0) {
    float a0 = 0.0f, a1 = 0.0f, a2 = 0.0f;
#pragma unroll
    for (int w = 0; w < NT / 32; ++w) { a0 += sv[w][0]; a1 += sv[w][1]; a2 += sv[w][2]; }
    part[blockIdx.x * 3 + 0] = a0;
    part[blockIdx.x * 3 + 1] = a1;
    part[blockIdx.x * 3 + 2] = a2;
  }
}

// ---------------------------------------------------------------------------
// Finalize: single block reduces NB block-partials and writes the two losses.
// ---------------------------------------------------------------------------
__global__ __launch_bounds__(256)
void av_loss_final(const float* __restrict__ part, float* __restrict__ out) {
  __shared__ float sh[256][3];
  float a0 = 0.0f, a1 = 0.0f, a2 = 0.0f;
  for (int i = threadIdx.x; i < NB; i += 256) {
    a0 += part[i * 3 + 0];
    a1 += part[i * 3 + 1];
    a2 += part[i * 3 + 2];
  }
  sh[threadIdx.x][0] = a0; sh[threadIdx.x][1] = a1; sh[threadIdx.x][2] = a2;
  __syncthreads();
  if (threadIdx.x == 0) {
    float s0 = 0.0f, s1 = 0.0f, s2 = 0.0f;
    for (int i = 0; i < 256; ++i) { s0 += sh[i][0]; s1 += sh[i][1]; s2 += sh[i][2]; }
    out[0] = s0 / (3.0f * s2);      // val_loss
    out[1] = 1.0f - s1 / s2;        // vec_loss
  }
}

extern "C" void kernel_launch(void* const* d_in, const int* in_sizes, int n_in,
                              void* d_out, int out_size, void* d_ws, size_t ws_size,
                              hipStream_t stream) {
  const float* in = (const float*)d_in[0];   // [2,6,80,80,80] f32
  const float* tg = (const float*)d_in[1];   // [2,6,80,80,80] f32
  const int*   mk = (const int*)d_in[2];     // [2,1,1,80,80,80] i32
  // d_in[3] = root_dir (unused by the reference forward)
  float* part = (float*)d_ws;                // NB*3 floats of scratch

  av_loss_main<<<NB, NT, 0, stream>>>(in, tg, mk, part);
  av_loss_final<<<1, 256, 0, stream>>>(part, (float*)d_out);
}